// PEBG_38826504356124
// MI455X (gfx1250) — compile-verified
//
#include <hip/hip_runtime.h>
#include <hip/hip_bf16.h>
#include <math.h>

typedef __attribute__((ext_vector_type(2))) float v2f;
typedef __attribute__((ext_vector_type(2))) int   v2i;
typedef __attribute__((ext_vector_type(8))) float v8f;
typedef int v4i __attribute__((vector_size(16)));   // matches builtin param type

#define BATCH    16384
#define NUM_TAGS 1024
#define DIM      64
#define HIDDEN   128
#define KC       256            // K-chunk for GEMM1 staging
#define NCHUNK   (NUM_TAGS / KC)
#define TS       260            // t_buf row stride (ints): bank-conflict-free A reads
#define ZS       200            // z row stride (floats), 16B-aligned rows
#define ES       130            // e row stride (floats)

// ---- gfx1250 async global->LDS path (ASYNCcnt-tracked), with sync fallback ----
#if defined(__has_builtin)
#  if __has_builtin(__builtin_amdgcn_global_load_async_to_lds_b128)
#    define HAVE_ASYNC_LDS 1
#  endif
#endif
#ifndef HAVE_ASYNC_LDS
#  define HAVE_ASYNC_LDS 0
#endif

typedef __attribute__((address_space(1))) v4i glob_v4i;   // global (AS1)
typedef __attribute__((address_space(3))) v4i lds_v4i;    // LDS (AS3)

__device__ __forceinline__ void wait_async_zero() {
#if HAVE_ASYNC_LDS
#  if defined(__has_builtin) && __has_builtin(__builtin_amdgcn_s_wait_asynccnt)
    __builtin_amdgcn_s_wait_asynccnt(0);
#  else
    asm volatile("s_wait_asynccnt 0x0" ::: "memory");
#  endif
#endif
}

__global__ __launch_bounds__(128)
void pebg_fused(const int*   __restrict__ questions,
                const int*   __restrict__ tgt,    // [B,1024] int32 0/1
                const float* __restrict__ dfeat,  // [B,3]
                const float* __restrict__ Qe,     // [NUM_Q,64]
                const float* __restrict__ Se,     // [1024,64]
                const float* __restrict__ Wd,     // [3,64]
                const float* __restrict__ bd,     // [64]
                const float* __restrict__ Wfc,    // [195,128]
                const float* __restrict__ bfc,    // [128]
                const float* __restrict__ Wout,   // [128]
                const float* __restrict__ bout,   // [1]
                float* __restrict__ eout,         // [B,128]
                float* __restrict__ pout)         // [B]
{
    __shared__ __align__(16) int   tibuf[2][16 * TS]; // staged t chunk (raw int32)
    __shared__ __align__(16) float zbuf[16 * ZS];     // z = [q | mu | a | p12 p13 p23 | 0]
    __shared__ float ebuf[16 * ES];
    __shared__ float cbuf[32];                        // per-half row counts

    const int tid  = threadIdx.x;
    const int lane = tid & 31;
    const int wave = tid >> 5;
    const int half = lane >> 4;      // 0: K+{0,1}, 1: K+{2,3} fragments of 16x16x4
    const int nl   = lane & 15;
    const int koff = half * 2;
    const int tile = blockIdx.x;     // 16 batch rows per block
    const long tbase = (long)tile * 16 * NUM_TAGS;

    // Warm L2 for the FC weights (gfx1250 global_prefetch_b8)
    __builtin_prefetch(Wfc + tid * 196, 0, 1);

    // ---- stage one 16x256 chunk of t into LDS (async DMA when available) ----
    auto stage = [&](int c, int buf) {
        const int* src = tgt + tbase + c * KC;
        #pragma unroll
        for (int j = 0; j < 8; ++j) {
            int elem = (j * 128 + tid) * 4;     // linear elem in 16x256 chunk
            int r  = elem >> 8;                 // /256
            int cc = elem & 255;
#if HAVE_ASYNC_LDS
            __builtin_amdgcn_global_load_async_to_lds_b128(
                (glob_v4i*)(src + (long)r * NUM_TAGS + cc),
                (lds_v4i*)&tibuf[buf][r * TS + cc],
                0, 0);
#else
            *(int4*)&tibuf[buf][r * TS + cc] = *(const int4*)(src + (long)r * NUM_TAGS + cc);
#endif
        }
    };

    // ================= GEMM1: mu_num = t @ S_emb via v_wmma_f32_16x16x4_f32 =====
    v8f acc = {0.f, 0.f, 0.f, 0.f, 0.f, 0.f, 0.f, 0.f};
    float cntf = 0.f;
    const int n = wave * 16 + nl;               // output column of mu (0..63)

    stage(0, 0);
    for (int c = 0; c < NCHUNK; ++c) {
        wait_async_zero();                      // drain our async stores to LDS
        __syncthreads();
        if (c + 1 < NCHUNK) stage(c + 1, (c + 1) & 1);   // double buffer next chunk
        const int* tb  = &tibuf[c & 1][nl * TS];         // A row = nl for both halves
        const int gk0  = c * KC;
        #pragma unroll 4
        for (int k = 0; k < KC; k += 4) {
            v2i ai = *(const v2i*)(tb + k + koff);
            v2f a;
            a.x = (float)ai.x;                  // t is exactly 0/1 -> exact f32
            a.y = (float)ai.y;
            v2f b;
            b.x = Se[(gk0 + k + koff)     * DIM + n];
            b.y = Se[(gk0 + k + koff + 1) * DIM + n];
            if (wave == 0) cntf += a.x + a.y;   // free row-count accumulation
            acc = __builtin_amdgcn_wmma_f32_16x16x4_f32(
                      false, a, false, b, (short)0, acc, false, false);
        }
    }

    if (wave == 0) cbuf[half * 16 + nl] = cntf;
    __syncthreads();

    // ---- mu = num / counts, written straight into z[:, 64:128] ----
    #pragma unroll
    for (int v = 0; v < 8; ++v) {
        int row   = v + 8 * half;                        // C/D layout: VGPR v, halves +8
        float cnt = cbuf[row] + cbuf[16 + row];
        zbuf[row * ZS + DIM + n] = acc[v] / cnt;
    }

    // ---- q gather into z[:, 0:64], a = df@W_diff + b into z[:, 128:192] ----
    {
        int r  = tid >> 3;
        int c0 = (tid & 7) * 8;
        int qi = questions[tile * 16 + r];
        const float* qrow = Qe + (long)qi * DIM + c0;
        *(float4*)&zbuf[r * ZS + c0]     = *(const float4*)(qrow);
        *(float4*)&zbuf[r * ZS + c0 + 4] = *(const float4*)(qrow + 4);
        float d0 = dfeat[(tile * 16 + r) * 3 + 0];
        float d1 = dfeat[(tile * 16 + r) * 3 + 1];
        float d2 = dfeat[(tile * 16 + r) * 3 + 2];
        #pragma unroll
        for (int j = 0; j < 8; ++j) {
            int d = c0 + j;
            zbuf[r * ZS + 2 * DIM + d] =
                fmaf(d0, Wd[d], fmaf(d1, Wd[64 + d], fmaf(d2, Wd[128 + d], bd[d])));
        }
    }
    __syncthreads();

    // ---- pairwise inner products p12/p13/p23 -> z[:, 192:195]; z[:,195] = 0 pad ----
    if (tid < 48) {
        int which = tid >> 4, r = tid & 15;
        int xa = (which == 2) ? DIM : 0;
        int xb = (which == 0) ? DIM : 2 * DIM;
        const float* zr = &zbuf[r * ZS];
        float s = 0.f;
        #pragma unroll 8
        for (int d = 0; d < DIM; ++d) s = fmaf(zr[xa + d], zr[xb + d], s);
        zbuf[r * ZS + 3 * DIM + which] = s;
    } else if (tid < 64) {
        zbuf[(tid - 48) * ZS + 195] = 0.f;               // K padded 195 -> 196
    }
    __syncthreads();

    // ================= GEMM2: e = relu(z @ W_fc + b_fc) via WMMA ================
    #pragma unroll
    for (int st = 0; st < 2; ++st) {
        int n2 = wave * 32 + st * 16 + nl;               // hidden column (0..127)
        float bias = bfc[n2];
        v8f acc2 = {bias, bias, bias, bias, bias, bias, bias, bias};
        const float* zr = &zbuf[nl * ZS];
        for (int k = 0; k < 196; k += 4) {
            v2f a = *(const v2f*)(zr + k + koff);
            int r0 = k + koff;  int r1 = r0 + 1;
            r0 = r0 > 194 ? 194 : r0;                    // clamp pad row (A is zero there)
            r1 = r1 > 194 ? 194 : r1;
            v2f b;
            b.x = Wfc[r0 * HIDDEN + n2];
            b.y = Wfc[r1 * HIDDEN + n2];
            acc2 = __builtin_amdgcn_wmma_f32_16x16x4_f32(
                       false, a, false, b, (short)0, acc2, false, false);
        }
        #pragma unroll
        for (int v = 0; v < 8; ++v) {
            int row = v + 8 * half;
            float e = fmaxf(acc2[v], 0.f);
            eout[((long)tile * 16 + row) * HIDDEN + n2] = e;
            ebuf[row * ES + n2] = e;
        }
    }
    __syncthreads();

    // ---- p = sigmoid(e @ W_out + b_out) ----
    if (tid < 16) {
        float s = bout[0];
        const float* er = &ebuf[tid * ES];
        #pragma unroll 8
        for (int h = 0; h < HIDDEN; ++h) s = fmaf(er[h], Wout[h], s);
        pout[tile * 16 + tid] = 1.f / (1.f + expf(-s));
    }
}

extern "C" void kernel_launch(void* const* d_in, const int* in_sizes, int n_in,
                              void* d_out, int out_size, void* d_ws, size_t ws_size,
                              hipStream_t stream) {
    const int*   questions = (const int*)  d_in[0];
    const int*   tgt       = (const int*)  d_in[1];
    const float* dfeat     = (const float*)d_in[2];
    const float* Qe        = (const float*)d_in[3];
    const float* Se        = (const float*)d_in[4];
    const float* Wd        = (const float*)d_in[5];
    const float* bd        = (const float*)d_in[6];
    const float* Wfc       = (const float*)d_in[7];
    const float* bfc       = (const float*)d_in[8];
    const float* Wout      = (const float*)d_in[9];
    const float* bout      = (const float*)d_in[10];

    float* eout = (float*)d_out;                         // e: [B,128] first
    float* pout = eout + (size_t)BATCH * HIDDEN;         // then p: [B]

    pebg_fused<<<BATCH / 16, 128, 0, stream>>>(
        questions, tgt, dfeat, Qe, Se, Wd, bd, Wfc, bfc, Wout, bout, eout, pout);
}